// HighOrderAggregator_24893630447801
// MI455X (gfx1250) — compile-verified
//
#include <hip/hip_runtime.h>
#include <hip/hip_bf16.h>

typedef float v2f __attribute__((ext_vector_type(2)));
typedef float v8f __attribute__((ext_vector_type(8)));

#define DIN 128
#define DOUT 128

// ---------------------------------------------------------------------------
// Kernel 1: zero hop1 workspace (float4 granularity)
// ---------------------------------------------------------------------------
__global__ void zero_f4(float4* __restrict__ p, int n4) {
    int i = blockIdx.x * blockDim.x + threadIdx.x;
    if (i < n4) p[i] = make_float4(0.f, 0.f, 0.f, 0.f);
}

// ---------------------------------------------------------------------------
// Kernel 2: SpMM scatter  hop1[rows[e]] += vals[e] * feat[cols[e]]
// One wave32 per edge; each lane owns 4 consecutive dims (float4).
// ---------------------------------------------------------------------------
__global__ __launch_bounds__(256) void spmm_scatter(
    const float* __restrict__ feat, const float* __restrict__ vals,
    const int* __restrict__ rows, const int* __restrict__ cols,
    float* __restrict__ hop1, int E)
{
    int e    = (blockIdx.x * blockDim.x + threadIdx.x) >> 5;
    int lane = threadIdx.x & 31;
    if (e >= E) return;
    int   r = rows[e];
    int   c = cols[e];
    float v = vals[e];
    float4 m = *((const float4*)(feat + (size_t)c * DIN) + lane);
    float* dst = hop1 + (size_t)r * DIN + lane * 4;
    atomicAdd(dst + 0, v * m.x);
    atomicAdd(dst + 1, v * m.y);
    atomicAdd(dst + 2, v * m.z);
    atomicAdd(dst + 3, v * m.w);
}

// ---------------------------------------------------------------------------
// Kernel 3: fused  out = norm(relu(feat@W0+b0)) + norm(relu(hop1@W1+b1))
// 8 waves / block, one 16-row tile per wave, fp32 WMMA 16x16x4.
// W staged in LDS pair-interleaved: Wp[k/2*128+n] = {W[k][n], W[k+1][n]}
// so each B fragment is a single aligned ds_load_b64 into an even VGPR pair.
// ---------------------------------------------------------------------------
__global__ __launch_bounds__(256) void fused_gemm_norm(
    const float* __restrict__ feat, const float* __restrict__ hop1,
    const float* __restrict__ W0, const float* __restrict__ b0,
    const float* __restrict__ off0, const float* __restrict__ sc0,
    const float* __restrict__ W1, const float* __restrict__ b1,
    const float* __restrict__ off1, const float* __restrict__ sc1,
    float* __restrict__ out, int n_tiles)
{
    __shared__ v2f Wp[(DIN / 2) * DOUT];   // 64 KB: pair-interleaved weights

    const int tid     = threadIdx.x;
    const int wave    = tid >> 5;
    const int lane    = tid & 31;
    const int lane_lo = lane & 15;
    const int half    = lane >> 4;         // 0: lanes 0-15, 1: lanes 16-31

    int tile = blockIdx.x * 8 + wave;
    if (tile >= n_tiles) tile = n_tiles - 1;   // duplicate work, identical stores
    const size_t row0 = (size_t)tile * 16;

    v8f res[8];
#pragma unroll
    for (int t = 0; t < 8; ++t) res[t] = (v8f)0.f;

#pragma unroll 1
    for (int phase = 0; phase < 2; ++phase) {
        const float* X  = phase ? hop1 : feat;
        const float* Wg = phase ? W1   : W0;
        const float* Bv = phase ? b1   : b0;
        const float* Ov = phase ? off1 : off0;
        const float* Sv = phase ? sc1  : sc0;

        // ---- stage W into LDS pair-interleaved (8192 pairs / 256 thr) -----
        __syncthreads();                       // prior phase done reading LDS
#pragma unroll
        for (int i = 0; i < 32; ++i) {
            int idx = tid + i * 256;           // 0..8191
            int k2  = idx >> 7;                // pair-row 0..63
            int n   = idx & 127;
            v2f p;
            p.x = Wg[(2 * k2)     * DOUT + n];
            p.y = Wg[(2 * k2 + 1) * DOUT + n];
            Wp[idx] = p;
        }
        __syncthreads();

        // ---- GEMM: 16 x 128 tile, K = 128 in steps of 4 -------------------
        v8f acc[8];
#pragma unroll
        for (int t = 0; t < 8; ++t) acc[t] = (v8f)0.f;

        // A frag: M = lane_lo, K pair = {k+2*half, k+2*half+1}
        const float* arow = X + (row0 + (size_t)lane_lo) * DIN + half * 2;

#pragma unroll 2
        for (int k = 0; k < DIN; k += 4) {
            v2f a = *(const v2f*)(arow + k);
            const int kb = ((k >> 1) + half) * DOUT + lane_lo;  // pair-row base
#pragma unroll
            for (int t = 0; t < 8; ++t) {
                v2f b = Wp[kb + t * 16];        // single ds_load_b64
                acc[t] = __builtin_amdgcn_wmma_f32_16x16x4_f32(
                    false, a, false, b, (short)0, acc[t], false, false);
            }
        }

        // ---- epilogue: bias + relu + row-wise norm ------------------------
        float bias_c[8], sc_c[8], off_c[8];
#pragma unroll
        for (int t = 0; t < 8; ++t) {
            int c = t * 16 + lane_lo;
            bias_c[t] = Bv[c];
            sc_c[t]   = Sv[c];
            off_c[t]  = Ov[c];
        }

        float s[8], q[8];
#pragma unroll
        for (int v = 0; v < 8; ++v) { s[v] = 0.f; q[v] = 0.f; }

#pragma unroll
        for (int t = 0; t < 8; ++t)
#pragma unroll
            for (int v = 0; v < 8; ++v) {
                float h = acc[t][v] + bias_c[t];
                h = fmaxf(h, 0.f);
                acc[t][v] = h;                   // keep relu'd value
                s[v] += h;
                q[v] += h * h;
            }

        // butterfly reduce across the 16-lane half that shares rows v+8*half
#pragma unroll
        for (int m = 1; m < 16; m <<= 1) {
#pragma unroll
            for (int v = 0; v < 8; ++v) {
                s[v] += __shfl_xor(s[v], m, 32);
                q[v] += __shfl_xor(q[v], m, 32);
            }
        }

        float mean[8], rstd[8];
#pragma unroll
        for (int v = 0; v < 8; ++v) {
            mean[v] = s[v] * (1.0f / DOUT);
            float var = q[v] * (1.0f / DOUT) - mean[v] * mean[v] + 1e-9f;
            rstd[v] = rsqrtf(var);
        }

#pragma unroll
        for (int t = 0; t < 8; ++t)
#pragma unroll
            for (int v = 0; v < 8; ++v)
                res[t][v] += (acc[t][v] - mean[v]) * sc_c[t] * rstd[v] + off_c[t];
    }

    // ---- store: row = row0 + v + 8*half, col = t*16 + lane_lo -------------
#pragma unroll
    for (int t = 0; t < 8; ++t)
#pragma unroll
        for (int v = 0; v < 8; ++v) {
            size_t r = row0 + v + 8 * half;
            out[r * DOUT + t * 16 + lane_lo] = res[t][v];
        }
}

// ---------------------------------------------------------------------------
extern "C" void kernel_launch(void* const* d_in, const int* in_sizes, int n_in,
                              void* d_out, int out_size, void* d_ws, size_t ws_size,
                              hipStream_t stream) {
    const float* feat = (const float*)d_in[0];
    const float* vals = (const float*)d_in[1];
    const float* W0   = (const float*)d_in[2];
    const float* b0   = (const float*)d_in[3];
    const float* off0 = (const float*)d_in[4];
    const float* sc0  = (const float*)d_in[5];
    const float* W1   = (const float*)d_in[6];
    const float* b1   = (const float*)d_in[7];
    const float* off1 = (const float*)d_in[8];
    const float* sc1  = (const float*)d_in[9];
    const int*   rows = (const int*)d_in[10];
    const int*   cols = (const int*)d_in[11];
    float*       out  = (float*)d_out;

    const int N = in_sizes[0] / DIN;   // 100000
    const int E = in_sizes[1];         // 3200000

    float* hop1 = (float*)d_ws;        // N*128 floats = 51.2 MB scratch

    // 1) zero hop1
    {
        int n4 = (N * DIN) / 4;
        int blocks = (n4 + 255) / 256;
        zero_f4<<<blocks, 256, 0, stream>>>((float4*)hop1, n4);
    }
    // 2) SpMM scatter (one wave per edge)
    {
        long long threads = (long long)E * 32;
        int blocks = (int)((threads + 255) / 256);
        spmm_scatter<<<blocks, 256, 0, stream>>>(feat, vals, rows, cols, hop1, E);
    }
    // 3) fused dual-GEMM + relu + rownorm + add
    {
        int n_tiles = N / 16;          // N divisible by 16
        int blocks = (n_tiles + 7) / 8;
        fused_gemm_norm<<<blocks, 256, 0, stream>>>(
            feat, hop1, W0, b0, off0, sc0, W1, b1, off1, sc1, out, n_tiles);
    }
}